// SliceTextureModule_28664611733894
// MI455X (gfx1250) — compile-verified
//
#include <hip/hip_runtime.h>

// ---------------------------------------------------------------------------
// Bilinear texture slice for MI455X (gfx1250, wave32).
//
// Memory-bound gather workload:
//   * 71 MB texture fits in the 192 MB L2 -> keep it resident with default RT
//     temporal policy on the gather loads.
//   * 272 MB output + 16 MB uv are streamed exactly once -> non-temporal
//     hints (CDNA5 TH bits) so they do NOT evict the texture from L2.
//   * wave32: 256-thread blocks = 8 waves; one sample per lane, fully
//     independent -> massive latency hiding for the divergent gathers.
// ---------------------------------------------------------------------------

namespace {

constexpr int TEX_H = 1024;
constexpr int TEX_W = 1024;
constexpr int TEX_C = 17;

// Naturally-aligned float4 (16B) and an under-aligned (4B) variant so the
// compiler may emit wide unaligned global_load_b128 for the 68-byte texel
// rows (gfx1250 supports unaligned global access).
typedef float f4 __attribute__((ext_vector_type(4)));
typedef float f4u __attribute__((ext_vector_type(4), aligned(4)));
typedef float f2 __attribute__((ext_vector_type(2)));

struct Tex17 {
    f4 a, b, c, d;  // channels 0..15
    float e;        // channel 16 (homogeneous)
};

__device__ __forceinline__ Tex17 load17(const float* __restrict__ p) {
    Tex17 r;
    const f4u* v = (const f4u*)p;
    r.a = v[0];
    r.b = v[1];
    r.c = v[2];
    r.d = v[3];
    r.e = p[16];
    return r;
}

// lerp expressed exactly like the reference: lo*(1-f) + hi*f
__device__ __forceinline__ Tex17 lerp17(const Tex17& lo, const Tex17& hi, float f) {
    const float g = 1.0f - f;
    Tex17 r;
    r.a = lo.a * g + hi.a * f;
    r.b = lo.b * g + hi.b * f;
    r.c = lo.c * g + hi.c * f;
    r.d = lo.d * g + hi.d * f;
    r.e = lo.e * g + hi.e * f;
    return r;
}

}  // namespace

__global__ __launch_bounds__(256) void slice_texture_kernel(
    const float* __restrict__ tex,   // [H, W, C]
    const float* __restrict__ uv,    // [N, 2]
    float* __restrict__ out,         // values[N,16] | homog[N,1] | vnn[N,17]
    int n_total) {
    const int n = blockIdx.x * blockDim.x + threadIdx.x;
    if (n >= n_total) return;

    // uv is streamed exactly once -> non-temporal load (don't pollute L2).
    const f2 u = __builtin_nontemporal_load((const f2*)uv + n);

    float x = u.x * (float)(TEX_W - 1);
    float y = u.y * (float)(TEX_H - 1);

    float xf = fminf(fmaxf(floorf(x), 0.0f), (float)(TEX_W - 2));
    float yf = fminf(fmaxf(floorf(y), 0.0f), (float)(TEX_H - 2));
    const int x0 = (int)xf;
    const int y0 = (int)yf;
    const float fx = x - xf;
    const float fy = y - yf;

    // Gather 4 texel rows (68 B each, L2-resident: default RT temporal hint).
    const float* p00 = tex + ((size_t)y0 * TEX_W + (size_t)x0) * TEX_C;
    const float* p10 = p00 + (size_t)TEX_W * TEX_C;

    const Tex17 t00 = load17(p00);
    const Tex17 t01 = load17(p00 + TEX_C);
    const Tex17 top = lerp17(t00, t01, fx);

    const Tex17 t10 = load17(p10);
    const Tex17 t11 = load17(p10 + TEX_C);
    const Tex17 bot = lerp17(t10, t11, fx);

    const Tex17 v = lerp17(top, bot, fy);

    const float h = v.e;
    const float inv = 1.0f / (h + 1e-5f);  // one IEEE divide, 16 multiplies

    // ---- outputs: all streamed once -> non-temporal stores (TH=NT) ----

    // values [N,16]: n*64 bytes -> always 16B aligned.
    f4* values = (f4*)(out + (size_t)n * 16);
    __builtin_nontemporal_store(v.a * inv, values + 0);
    __builtin_nontemporal_store(v.b * inv, values + 1);
    __builtin_nontemporal_store(v.c * inv, values + 2);
    __builtin_nontemporal_store(v.d * inv, values + 3);

    // homogeneous [N,1]
    float* homog = out + (size_t)n_total * 16;
    __builtin_nontemporal_store(h, homog + n);

    // values_not_normalized [N,17]: base is 4B aligned only.
    float* vnn = out + (size_t)n_total * 17 + (size_t)n * 17;
    __builtin_nontemporal_store(v.a, (f4u*)vnn + 0);
    __builtin_nontemporal_store(v.b, (f4u*)vnn + 1);
    __builtin_nontemporal_store(v.c, (f4u*)vnn + 2);
    __builtin_nontemporal_store(v.d, (f4u*)vnn + 3);
    __builtin_nontemporal_store(v.e, vnn + 16);
}

extern "C" void kernel_launch(void* const* d_in, const int* in_sizes, int n_in,
                              void* d_out, int out_size, void* d_ws, size_t ws_size,
                              hipStream_t stream) {
    const float* tex = (const float*)d_in[0];   // [1024,1024,17] f32
    const float* uv  = (const float*)d_in[1];   // [N,2] f32
    float* out = (float*)d_out;

    const int n = in_sizes[1] / 2;              // N samples
    constexpr int kThreads = 256;               // 8 wave32 waves per block
    const int blocks = (n + kThreads - 1) / kThreads;

    slice_texture_kernel<<<blocks, kThreads, 0, stream>>>(tex, uv, out, n);
}